// LDCA_4690104287721
// MI455X (gfx1250) — compile-verified
//
#include <hip/hip_runtime.h>

typedef __attribute__((ext_vector_type(2))) float v2f;
typedef __attribute__((ext_vector_type(8))) float v8f;

#define HWN   65536     // H*W
#define CIN   64
#define C2    32
#define C4    16
#define NBAT  16
#define LLEN  4096      // (H/4)*(W/4)

// Store with TH_STORE_WB: keep the line dirty in L2 (no write-rinse to HBM).
// Used for the x_ intermediate, which is L2-resident (128 MiB < 192 MB L2).
__device__ inline void store_f32_wb(float* p, float v) {
  asm volatile("global_store_b32 %0, %1, off th:TH_STORE_WB"
               :: "v"(p), "v"(v)
               : "memory");
}

// ---------------------------------------------------------------------------
// K1: x_ = relu(w1 @ x + b1)   per batch: (32x64)@(64x65536)
// One wave -> 32 out-channels x 16 pixels, via 16 K-steps of wmma f32 16x16x4.
// x is streamed with NT loads (read once; don't evict x_ from L2).
// ---------------------------------------------------------------------------
__global__ __launch_bounds__(256)
void k1_gemm1(const float* __restrict__ x, const float* __restrict__ w1,
              const float* __restrict__ b1, float* __restrict__ xp) {
  const int lane = threadIdx.x & 31;
  const int wave = threadIdx.x >> 5;
  const int b    = blockIdx.y;
  const int n0   = (blockIdx.x * 8 + wave) * 16;
  const int hi   = lane >> 4;       // 0: lanes 0-15 (K 0,1) ; 1: lanes 16-31 (K 2,3)
  const int ln   = lane & 15;

  // A fragments (w1 rows, held in registers for the whole wave)
  v2f a[2][16];
#pragma unroll
  for (int mt = 0; mt < 2; ++mt) {
    const int m = mt * 16 + ln;
#pragma unroll
    for (int ks = 0; ks < 16; ++ks) {
      const int k0 = ks * 4 + hi * 2;
      a[mt][ks] = *(const v2f*)(w1 + m * CIN + k0);
    }
  }

  const float* xb = x + (size_t)b * CIN * HWN;
  v8f acc0 = {};
  v8f acc1 = {};
#pragma unroll
  for (int ks = 0; ks < 16; ++ks) {
    const int k0 = ks * 4 + hi * 2;
    v2f bf;
    bf.x = __builtin_nontemporal_load(&xb[(size_t)(k0    ) * HWN + n0 + ln]);
    bf.y = __builtin_nontemporal_load(&xb[(size_t)(k0 + 1) * HWN + n0 + ln]);
    acc0 = __builtin_amdgcn_wmma_f32_16x16x4_f32(false, a[0][ks], false, bf,
                                                 (short)0, acc0, false, false);
    acc1 = __builtin_amdgcn_wmma_f32_16x16x4_f32(false, a[1][ks], false, bf,
                                                 (short)0, acc1, false, false);
  }

  float* xpb = xp + (size_t)b * C2 * HWN;
#pragma unroll
  for (int r = 0; r < 8; ++r) {
    const int m0 = r + hi * 8;                       // D layout: m = vgpr + 8*hi
    const float v0 = fmaxf(acc0[r] + b1[m0], 0.f);
    const float v1 = fmaxf(acc1[r] + b1[16 + m0], 0.f);
    store_f32_wb(&xpb[(size_t)(m0     ) * HWN + n0 + ln], v0);
    store_f32_wb(&xpb[(size_t)(16 + m0) * HWN + n0 + ln], v1);
  }
}

// ---------------------------------------------------------------------------
// K2: per (b,ch) compute sum, sum^2, sum^3 over the 4096 positions of each of
// the 16 block-slots s = (h%4)*4 + (w%4).  Coalesced: thread t streams
// n = t + 256*i  ->  w%4 = t&3 (const), h%4 = i&3 (cycles).
// Reads x_ with regular RT loads -> L2 hits.
// ---------------------------------------------------------------------------
__global__ __launch_bounds__(256)
void k2_stats(const float* __restrict__ xp, float* __restrict__ stats) {
  const int ch = blockIdx.x;       // 0..31
  const int b  = blockIdx.y;
  const int t  = threadIdx.x;
  const float* p = xp + ((size_t)b * C2 + ch) * HWN;

  float s1[4] = {0.f, 0.f, 0.f, 0.f};
  float s2[4] = {0.f, 0.f, 0.f, 0.f};
  float s3[4] = {0.f, 0.f, 0.f, 0.f};
  for (int i = 0; i < 256; ++i) {
    const float v  = p[t + 256 * i];
    const float vv = v * v;
    const int hh = i & 3;
    s1[hh] += v;
    s2[hh] += vv;
    s3[hh] += vv * v;
  }

  __shared__ float red[48];        // 16 slots * 3 stats
  if (t < 48) red[t] = 0.f;
  __syncthreads();
  const int wmod = t & 3;
#pragma unroll
  for (int hh = 0; hh < 4; ++hh) {
    const int s = hh * 4 + wmod;
    atomicAdd(&red[s * 3 + 0], s1[hh]);
    atomicAdd(&red[s * 3 + 1], s2[hh]);
    atomicAdd(&red[s * 3 + 2], s3[hh]);
  }
  __syncthreads();
  if (t < 48) stats[(((size_t)b * C2 + ch) * 16) * 3 + t] = red[t];
}

// ---------------------------------------------------------------------------
// K3: qkv = mean + m3 ; per-channel 16x16 softmax attention ; sigmoid gate.
// One block per batch.
// ---------------------------------------------------------------------------
__device__ inline float qkv_from(const float* s) {
  const float s1 = s[0], s2 = s[1], s3 = s[2];
  const float mean = s1 * (1.f / 4096.f);
  const float m3 = (s3 - 3.f * mean * s2 + 3.f * mean * mean * s1
                    - (float)LLEN * mean * mean * mean) * (1.f / 16.f);
  return mean + m3;
}

__global__ __launch_bounds__(256)
void k3_attn(const float* __restrict__ stats, float* __restrict__ sig) {
  const int b = blockIdx.x;
  const int t = threadIdx.x;
  __shared__ float qm[256];        // q[ch][s], ch 0..15
  __shared__ float vm[256];        // v[ch][s] (qkv channels 16..31)
  {
    const int ch = t >> 4, s = t & 15;
    const float* sq = stats + (((size_t)b * C2 + ch     ) * 16 + s) * 3;
    const float* sv = stats + (((size_t)b * C2 + ch + 16) * 16 + s) * 3;
    qm[t] = qkv_from(sq);
    vm[t] = qkv_from(sv);
  }
  __syncthreads();

  const int ch = t >> 4, i = t & 15;
  const float qi = qm[ch * 16 + i];
  float mx = -3.402823466e38f;
#pragma unroll
  for (int j = 0; j < 16; ++j) mx = fmaxf(mx, qi * qm[ch * 16 + j]);
  float den = 0.f, num = 0.f;
#pragma unroll
  for (int j = 0; j < 16; ++j) {
    const float e = __expf(qi * qm[ch * 16 + j] - mx);
    den += e;
    num += e * vm[ch * 16 + j];
  }
  const float y = num / den;
  sig[b * 256 + t] = 1.f / (1.f + __expf(-y));
}

// ---------------------------------------------------------------------------
// K4: out = w2 @ (vx * sig) + b2 + x   per batch: (64x16)@(16x65536)
// One wave -> 64 out-channels x 16 pixels, 4 K-steps of wmma f32 16x16x4.
// vx read RT (L2 hit); residual x read NT; out stored NT (never re-read).
// ---------------------------------------------------------------------------
__global__ __launch_bounds__(256)
void k4_gemm2(const float* __restrict__ xp, const float* __restrict__ x,
              const float* __restrict__ w2, const float* __restrict__ b2,
              const float* __restrict__ sig, float* __restrict__ out) {
  const int lane = threadIdx.x & 31;
  const int wave = threadIdx.x >> 5;
  const int b    = blockIdx.y;
  const int n0   = (blockIdx.x * 8 + wave) * 16;
  const int hi   = lane >> 4;
  const int ln   = lane & 15;

  __shared__ float sg[256];                 // sig[ch][s] for this batch
  sg[threadIdx.x] = sig[b * 256 + threadIdx.x];
  __syncthreads();

  v2f a[4][4];                              // w2 (64x16) A-fragments
#pragma unroll
  for (int mt = 0; mt < 4; ++mt) {
    const int m = mt * 16 + ln;
#pragma unroll
    for (int ks = 0; ks < 4; ++ks)
      a[mt][ks] = *(const v2f*)(w2 + m * C4 + ks * 4 + hi * 2);
  }

  const float* vxb = xp + ((size_t)b * C2 + C4) * HWN;  // channels 16..31 of x_
  const int n = n0 + ln;
  const int s = (((n0 >> 8) & 3) << 2) | (ln & 3);      // (h%4)*4 + (w%4)

  v8f acc[4] = {};
#pragma unroll
  for (int ks = 0; ks < 4; ++ks) {
    const int c0 = ks * 4 + hi * 2;
    v2f bf;
    bf.x = vxb[(size_t)(c0    ) * HWN + n] * sg[(c0    ) * 16 + s];
    bf.y = vxb[(size_t)(c0 + 1) * HWN + n] * sg[(c0 + 1) * 16 + s];
#pragma unroll
    for (int mt = 0; mt < 4; ++mt)
      acc[mt] = __builtin_amdgcn_wmma_f32_16x16x4_f32(false, a[mt][ks], false, bf,
                                                      (short)0, acc[mt], false, false);
  }

  const float* xres = x + (size_t)b * CIN * HWN;
  float* ob = out + (size_t)b * CIN * HWN;
#pragma unroll
  for (int mt = 0; mt < 4; ++mt) {
#pragma unroll
    for (int r = 0; r < 8; ++r) {
      const int m = mt * 16 + r + 8 * hi;
      const float res = __builtin_nontemporal_load(&xres[(size_t)m * HWN + n]);
      __builtin_nontemporal_store(acc[mt][r] + b2[m] + res,
                                  &ob[(size_t)m * HWN + n]);
    }
  }
}

// ---------------------------------------------------------------------------
extern "C" void kernel_launch(void* const* d_in, const int* in_sizes, int n_in,
                              void* d_out, int out_size, void* d_ws, size_t ws_size,
                              hipStream_t stream) {
  const float* x  = (const float*)d_in[0];
  const float* w1 = (const float*)d_in[1];
  const float* b1 = (const float*)d_in[2];
  const float* w2 = (const float*)d_in[3];
  const float* b2 = (const float*)d_in[4];
  float* out = (float*)d_out;

  float* xp    = (float*)d_ws;                                    // 16*32*65536*4 = 128 MiB
  float* stats = (float*)((char*)d_ws + (size_t)NBAT * C2 * HWN * sizeof(float));
  float* sig   = stats + (size_t)NBAT * C2 * 16 * 3;              // 96 KiB after xp

  k1_gemm1<<<dim3(512, NBAT), 256, 0, stream>>>(x, w1, b1, xp);
  k2_stats<<<dim3(C2, NBAT), 256, 0, stream>>>(xp, stats);
  k3_attn<<<NBAT, 256, 0, stream>>>(stats, sig);
  k4_gemm2<<<dim3(512, NBAT), 256, 0, stream>>>(xp, x, w2, b2, sig, out);
}